// FWI_24910810317219
// MI455X (gfx1250) — compile-verified
//
#include <hip/hip_runtime.h>
#include <hip/hip_bf16.h>

// ---------------- problem constants (match reference) ----------------
#define NZ      300
#define NX      400
#define NPML    32
#define NZP     364                 // NZ + 2*NPML
#define NXP     464                 // NX + 2*NPML
#define NSTEPS  200
#define NSHOTS  2
#define DT      0.001f
#define DXI     100.0f              // 1/DX, DX = 0.01
#define SRC_Z   34                  // NPML + 2
#define REC_Z   34                  // NPML + 2
#define NEL     (NZP * NXP)         // 168,896 interior cells
// haloed plane: 1-cell permanently-zero ring implements the zero-pad
// boundary semantics of _dxf/_dxb/_dzf/_dzb with no conditionals.
#define HZP     (NZP + 2)           // 366
#define HXP     (NXP + 2)           // 466
#define HEL     (HZP * HXP)         // 170,556
#define NREC    (NSHOTS * NSTEPS * NX)   // 160,000 (multiple of 64)
#define R1_BLOCKS 64

// stencil block tiling
#define BX 32
#define BZ 8
#define TX (BX + 2)                 // 34 (tile width, elements)
#define TZ (BZ + 2)                 // 10 (tile height, rows)

typedef float v2f __attribute__((ext_vector_type(2)));
typedef float v8f __attribute__((ext_vector_type(8)));
typedef unsigned int u32x4 __attribute__((ext_vector_type(4)));
typedef int i32x4 __attribute__((ext_vector_type(4)));
typedef int i32x8 __attribute__((ext_vector_type(8)));

// ---- Tensor Data Mover: stage a TZ x TX f32 tile of a haloed plane into LDS.
// D# per CDNA5 ISA: group0 = {count/lds_addr/global_addr/type},
// group1 = {data_size, tensor_dim0/1, tile_dim0/1, tensor_dim0_stride}.
// (z0,x0) are haloed-plane coords of the tile origin (always >= 0 thanks to halo);
// reads past tensor_dim return zero (only lands in unused tile slots).
// This toolchain exposes the 6-arg builtin:
//   (u32x4 g0, i32x8 g1, i32x4 g2, i32x4 g3, i32x8 extra, i32 cpol)
__device__ __forceinline__ void tdm_tile_load(const float* plane, int z0, int x0,
                                              void* lds_dst) {
  unsigned long long ga = (unsigned long long)(uintptr_t)plane +
                          (unsigned long long)((unsigned)(z0 * HXP + x0)) * 4ull;
  unsigned int lds_off = (unsigned int)(uintptr_t)lds_dst;  // low 32 bits = LDS offset
  u32x4 g0;
  g0[0] = 1u;                                            // count=1, user mode
  g0[1] = lds_off;                                       // lds_addr (bytes)
  g0[2] = (unsigned int)(ga & 0xffffffffull);            // global_addr[31:0]
  g0[3] = (unsigned int)((ga >> 32) & 0x01ffffffull)     // global_addr[56:32]
          | (2u << 30);                                  // type = 2 ("image")
  i32x8 g1;
  g1[0] = (int)(2u << 16);                               // wg_mask=0, data_size=2 (4B)
  g1[1] = (int)(((unsigned)HXP & 0xffffu) << 16);        // tensor_dim0[15:0]
  g1[2] = (int)(((unsigned)HZP & 0xffffu) << 16);        // dim0[31:16]=0 | tensor_dim1[15:0]
  g1[3] = (int)(((unsigned)TX & 0xffffu) << 16);         // dim1[31:16]=0 | tile_dim0
  g1[4] = (int)((unsigned)TZ & 0xffffu);                 // tile_dim1 | tile_dim2=0
  g1[5] = (int)HXP;                                      // tensor_dim0_stride[31:0]
  g1[6] = 0;                                             // stride0 hi | stride1 lo (2D: unused)
  g1[7] = 0;
  i32x4 z4 = {0, 0, 0, 0};                               // groups 2/3 unused (2D)
  i32x8 z8 = {0, 0, 0, 0, 0, 0, 0, 0};                   // unused trailing group
  __builtin_amdgcn_tensor_load_to_lds(g0, g1, z4, z4, z8, /*cpol=*/0);
}

// ---------------- setup: materials, sponge, zero state ----------------
// Forward value of Mask*X + (1-Mask)*stop_grad(X) is X, so Mask is unused.
__global__ void fwi_setup(const float* __restrict__ Vp,
                          const float* __restrict__ Vs,
                          const float* __restrict__ Den,
                          float* __restrict__ lam, float* __restrict__ mu,
                          float* __restrict__ lam2mu, float* __restrict__ dtrho,
                          float* __restrict__ damp, float* __restrict__ fields) {
  int gid = blockIdx.x * blockDim.x + threadIdx.x;
  if (gid >= HEL) return;
  // zero all 10 haloed state planes (vx/vz/sxx/szz/sxz x 2 shots), incl. halo ring
#pragma unroll
  for (int p = 0; p < 10; ++p) fields[p * HEL + gid] = 0.0f;
  int izh = gid / HXP;
  int jxh = gid - izh * HXP;
  int iz = izh - 1, jx = jxh - 1;
  if (iz < 0 || iz >= NZP || jx < 0 || jx >= NXP) return;   // halo: state stays 0
  int m = iz * NXP + jx;
  // edge-mode padding of the material model
  int i0 = min(max(iz - NPML, 0), NZ - 1);
  int j0 = min(max(jx - NPML, 0), NX - 1);
  float vp = Vp[i0 * NX + j0];
  float vs = Vs[i0 * NX + j0];
  float dn = Den[i0 * NX + j0];
  float muv  = vs * vs * dn * 1.0e-6f;
  float lamv = (vp * vp - 2.0f * vs * vs) * dn * 1.0e-6f;
  lam[m]    = lamv;
  mu[m]     = muv;
  lam2mu[m] = lamv + 2.0f * muv;
  dtrho[m]  = DT / dn;
  float dz = fminf(fmaxf(fmaxf((float)(NPML - iz), (float)(iz - (NZP - 1 - NPML))), 0.0f),
                   (float)NPML) * (1.0f / (float)NPML);
  float dxq = fminf(fmaxf(fmaxf((float)(NPML - jx), (float)(jx - (NXP - 1 - NPML))), 0.0f),
                    (float)NPML) * (1.0f / (float)NPML);
  damp[m] = expf(-0.1f * (dz * dz + dxq * dxq));
}

// ---------------- velocity update: vx,vz <- sxx,szz,sxz ----------------
__global__ void fwi_vel(const float* __restrict__ dtrho,
                        const float* __restrict__ damp,
                        float* __restrict__ fields) {
  __shared__ float t_sxx[TZ * TX];
  __shared__ float t_szz[TZ * TX];
  __shared__ float t_sxz[TZ * TX];
  int shot = blockIdx.z;
  int jx0 = blockIdx.x * BX;
  int iz0 = blockIdx.y * BZ;
  const float* sxx = fields + (4 + shot) * HEL;
  const float* szz = fields + (6 + shot) * HEL;
  const float* sxz = fields + (8 + shot) * HEL;
  // wave 0 stages 3 tiles via the Tensor Data Mover, then waits TENSORcnt.
  if (threadIdx.y == 0) {
    tdm_tile_load(sxx, iz0, jx0, t_sxx);   // haloed origin (iz0, jx0) = interior (iz0-1, jx0-1)
    tdm_tile_load(szz, iz0, jx0, t_szz);
    tdm_tile_load(sxz, iz0, jx0, t_sxz);
    __builtin_amdgcn_s_wait_tensorcnt(0);
  }
  __syncthreads();
  int tx = threadIdx.x, tz = threadIdx.y;
  int jx = jx0 + tx, iz = iz0 + tz;
  if (jx >= NXP || iz >= NZP) return;
  float sxx_c = t_sxx[(tz + 1) * TX + (tx + 1)];
  float sxx_r = t_sxx[(tz + 1) * TX + (tx + 2)];   // dxf
  float sxz_c = t_sxz[(tz + 1) * TX + (tx + 1)];
  float sxz_u = t_sxz[(tz    ) * TX + (tx + 1)];   // dzb
  float sxz_l = t_sxz[(tz + 1) * TX + (tx    )];   // dxb
  float szz_c = t_szz[(tz + 1) * TX + (tx + 1)];
  float szz_d = t_szz[(tz + 2) * TX + (tx + 1)];   // dzf
  int m = iz * NXP + jx;
  int h = (iz + 1) * HXP + (jx + 1);
  float dr = dtrho[m];
  float dp = damp[m];
  float* vx = fields + (0 + shot) * HEL;
  float* vz = fields + (2 + shot) * HEL;
  vx[h] = (vx[h] + dr * ((sxx_r - sxx_c) + (sxz_c - sxz_u)) * DXI) * dp;
  vz[h] = (vz[h] + dr * ((sxz_c - sxz_l) + (szz_d - szz_c)) * DXI) * dp;
}

// ------- stress update + source injection + receiver recording --------
__global__ void fwi_stress(const float* __restrict__ lam,
                           const float* __restrict__ mu,
                           const float* __restrict__ lam2mu,
                           const float* __restrict__ damp,
                           float* __restrict__ fields,
                           const float* __restrict__ Stf,
                           const int* __restrict__ shot_ids,
                           float* __restrict__ recs, int t) {
  __shared__ float t_vx[TZ * TX];
  __shared__ float t_vz[TZ * TX];
  int shot = blockIdx.z;
  int jx0 = blockIdx.x * BX;
  int iz0 = blockIdx.y * BZ;
  const float* vx = fields + (0 + shot) * HEL;
  const float* vz = fields + (2 + shot) * HEL;
  if (threadIdx.y == 0) {
    tdm_tile_load(vx, iz0, jx0, t_vx);
    tdm_tile_load(vz, iz0, jx0, t_vz);
    __builtin_amdgcn_s_wait_tensorcnt(0);
  }
  __syncthreads();
  int tx = threadIdx.x, tz = threadIdx.y;
  int jx = jx0 + tx, iz = iz0 + tz;
  if (jx >= NXP || iz >= NZP) return;
  float vx_c = t_vx[(tz + 1) * TX + (tx + 1)];
  float vx_l = t_vx[(tz + 1) * TX + (tx    )];     // dxb
  float vx_d = t_vx[(tz + 2) * TX + (tx + 1)];     // dzf
  float vz_c = t_vz[(tz + 1) * TX + (tx + 1)];
  float vz_u = t_vz[(tz    ) * TX + (tx + 1)];     // dzb
  float vz_r = t_vz[(tz + 1) * TX + (tx + 2)];     // dxf
  float dvx_dx = (vx_c - vx_l) * DXI;
  float dvz_dz = (vz_c - vz_u) * DXI;
  int m = iz * NXP + jx;
  int h = (iz + 1) * HXP + (jx + 1);
  float dp  = damp[m];
  float l   = lam[m];
  float l2m = lam2mu[m];
  float mm  = mu[m];
  float* sxx = fields + (4 + shot) * HEL;
  float* szz = fields + (6 + shot) * HEL;
  float* sxz = fields + (8 + shot) * HEL;
  float nsxx = (sxx[h] + DT * (l2m * dvx_dx + l   * dvz_dz)) * dp;
  float nszz = (szz[h] + DT * (l   * dvx_dx + l2m * dvz_dz)) * dp;
  float nsxz = (sxz[h] + DT * mm * ((vx_d - vx_c) + (vz_r - vz_c)) * DXI) * dp;
  // source injection (after damping, matching .at[].add in reference)
  int sid   = shot_ids[shot];
  int src_x = NPML + 20 + sid * ((NX - 40) / NSHOTS);
  if (iz == SRC_Z && jx == src_x) {
    float s = Stf[sid * NSTEPS + t] * DT;
    nsxx += s;
    nszz += s;
  }
  sxx[h] = nsxx;
  szz[h] = nszz;
  sxz[h] = nsxz;
  // record vx (post-velocity-update state; stress doesn't modify vx)
  if (iz == REC_Z && jx >= NPML && jx < NPML + NX) {
    recs[(shot * NSTEPS + t) * NX + (jx - NPML)] = vx_c;
  }
}

// ---- stage-1 reduction: sum of squares via V_WMMA_F32_16X16X4_F32 ----
// B = all-ones => D[m][n] = sum_k A[m][k], accumulated in C; a total-sum is
// invariant to the A lane permutation. NREC % 64 == 0 and base % 64 == 0,
// so no bounds guards are needed -> tight load/square/wmma loop, EXEC full.
__global__ void fwi_reduce1(const float* __restrict__ recs,
                            float* __restrict__ partials) {
  const int NTOT = NREC;
  int lane          = threadIdx.x & 31;
  int wave_in_blk   = threadIdx.x >> 5;
  int waves_per_blk = blockDim.x >> 5;
  int wave_global   = blockIdx.x * waves_per_blk + wave_in_blk;
  int nwaves        = gridDim.x * waves_per_blk;
  v2f ones; ones[0] = 1.0f; ones[1] = 1.0f;
  v8f c = {};
  for (int base = wave_global * 64; base < NTOT; base += nwaves * 64) {
    float x0 = recs[base + lane];
    float x1 = recs[base + 32 + lane];
    v2f a; a[0] = x0 * x0; a[1] = x1 * x1;
    c = __builtin_amdgcn_wmma_f32_16x16x4_f32(
        /*neg_a=*/false, a, /*neg_b=*/false, ones,
        /*c_mod=*/(short)0, c, /*reuse_a=*/false, /*reuse_b=*/false);
  }
  float s = c[0] + c[1] + c[2] + c[3] + c[4] + c[5] + c[6] + c[7];
  float wave_sum = s + __shfl_xor(s, 16, 32);   // rows 0-7 + rows 8-15
  __shared__ float lds[8];
  if (lane == 0) lds[wave_in_blk] = wave_sum;
  __syncthreads();
  if (threadIdx.x == 0) {
    float tot = 0.0f;
    for (int w = 0; w < waves_per_blk; ++w) tot += lds[w];
    partials[blockIdx.x] = tot;
  }
}

// ---- stage-2 reduction: fixed-order (deterministic across replays) ----
__global__ void fwi_reduce2(const float* __restrict__ partials,
                            float* __restrict__ out) {
  if (blockIdx.x == 0 && threadIdx.x == 0) {
    float tot = 0.0f;
    for (int i = 0; i < R1_BLOCKS; ++i) tot += partials[i];
    out[0] = 0.5f * tot;
  }
}

// ----------------------------- launcher -------------------------------
extern "C" void kernel_launch(void* const* d_in, const int* in_sizes, int n_in,
                              void* d_out, int out_size, void* d_ws, size_t ws_size,
                              hipStream_t stream) {
  const float* Vp       = (const float*)d_in[0];
  const float* Vs       = (const float*)d_in[1];
  const float* Den      = (const float*)d_in[2];
  const float* Stf      = (const float*)d_in[3];
  // d_in[4] = Mask: dead in the forward pass (stop_gradient identity)
  const int*   Shot_ids = (const int*)d_in[5];
  // d_in[6] = ngpu scalar: unused

  float* ws       = (float*)d_ws;
  float* lam      = ws;                     // NEL
  float* mu_      = lam + NEL;              // NEL
  float* lam2mu   = mu_ + NEL;              // NEL
  float* dtrho    = lam2mu + NEL;           // NEL
  float* damp     = dtrho + NEL;            // NEL
  float* fields   = damp + NEL;             // 10 * HEL (haloed vx,vz,sxx,szz,sxz x 2)
  float* recs     = fields + 10 * HEL;      // NREC
  float* partials = recs + NREC;            // R1_BLOCKS
  // total ws use: (5*NEL + 10*HEL + NREC + 64) * 4  ~= 10.9 MB (L2-resident)

  fwi_setup<<<(HEL + 255) / 256, 256, 0, stream>>>(Vp, Vs, Den, lam, mu_, lam2mu,
                                                   dtrho, damp, fields);
  dim3 blk(BX, BZ, 1);
  dim3 grd((NXP + BX - 1) / BX, (NZP + BZ - 1) / BZ, NSHOTS);
  for (int t = 0; t < NSTEPS; ++t) {
    fwi_vel<<<grd, blk, 0, stream>>>(dtrho, damp, fields);
    fwi_stress<<<grd, blk, 0, stream>>>(lam, mu_, lam2mu, damp, fields,
                                        Stf, Shot_ids, recs, t);
  }
  fwi_reduce1<<<R1_BLOCKS, 256, 0, stream>>>(recs, partials);
  fwi_reduce2<<<1, 32, 0, stream>>>(partials, (float*)d_out);
}